// SmallLM_17033840296517
// MI455X (gfx1250) — compile-verified
//
#include <hip/hip_runtime.h>
#include <hip/hip_bf16.h>
#include <math.h>
#include <stdint.h>

#define LL 6
#define HH 8
#define DD 64
#define EE 512
#define VV 32000
#define SSQ 2048
#define BBB 2
#define MM (BBB * SSQ) /* 4096 rows */

typedef float v2f __attribute__((ext_vector_type(2)));
typedef float v8f __attribute__((ext_vector_type(8)));

// D = A(16x4) * B(4x16) + C, fp32, wave32. A: lane m=lane%16, holds K=kb,kb+1
// (kb = 2*(lane/16)). B/C/D: rows striped across lanes within a VGPR.
__device__ __forceinline__ v8f wmma4(v2f a, v2f b, v8f c) {
  return __builtin_amdgcn_wmma_f32_16x16x4_f32(false, a, false, b, (short)0, c,
                                               false, false);
}

__device__ __forceinline__ v8f zero8() {
  v8f z = {0.f, 0.f, 0.f, 0.f, 0.f, 0.f, 0.f, 0.f};
  return z;
}

// CDNA5 async DMA: global -> LDS, 16B per lane, tracked by ASYNCcnt.
#define ASYNC_B128(ldsOff, gaddr)                                          \
  asm volatile("global_load_async_to_lds_b128 %0, %1, off" ::"v"(ldsOff), \
               "v"(gaddr)                                                  \
               : "memory")
#define WAIT_ASYNC() asm volatile("s_wait_asynccnt 0" ::: "memory")

__device__ __forceinline__ unsigned lds_off(const void* p) {
  return (unsigned)(uintptr_t)p; // flat LDS addr: offset lives in addr[31:0]
}

// ---------------------------------------------------------------- embedding
__global__ __launch_bounds__(256) void embed_k(const int* __restrict__ ids,
                                               const float* __restrict__ tok,
                                               const float* __restrict__ pos,
                                               float* __restrict__ x) {
  const int idx = blockIdx.x * 256 + threadIdx.x; // over MM*EE
  const int row = idx >> 9;                       // token position (b*S+s)
  const int col = idx & (EE - 1);
  const int s = row & (SSQ - 1);
  x[idx] = tok[(size_t)ids[row] * EE + col] * 0.0441941738241592f +
           pos[(size_t)s * EE + col];
}

// ---------------------------------------------------------------- layernorm
// one wave per row of 512; 16 elements per lane; shfl-xor reductions
__global__ __launch_bounds__(256) void ln_rows(const float* __restrict__ x,
                                               const float* __restrict__ g,
                                               float* __restrict__ y) {
  const int lane = threadIdx.x & 31;
  const int w = threadIdx.x >> 5;
  const int row = blockIdx.x * 8 + w;
  const float* xr = x + (size_t)row * EE;
  float v[16];
  float s = 0.0f;
#pragma unroll
  for (int i = 0; i < 16; ++i) {
    v[i] = xr[lane + i * 32];
    s += v[i];
  }
#pragma unroll
  for (int off = 16; off >= 1; off >>= 1) s += __shfl_xor(s, off);
  const float mu = s * (1.0f / EE);
  float var = 0.0f;
#pragma unroll
  for (int i = 0; i < 16; ++i) {
    const float d = v[i] - mu;
    var += d * d;
  }
#pragma unroll
  for (int off = 16; off >= 1; off >>= 1) var += __shfl_xor(var, off);
  const float rstd = rsqrtf(var * (1.0f / EE) + 1e-6f);
  float* yr = y + (size_t)row * EE;
#pragma unroll
  for (int i = 0; i < 16; ++i) {
    const int c = lane + i * 32;
    yr[c] = (v[i] - mu) * rstd * g[c];
  }
}

// ---------------------------------------------------------------- GEMM
// C[M,N] = epi(A[M,K] @ B[K,N] + bias [+ Res]).  EPI: 0=bias, 1=bias+GELU,
// 2=bias+residual.  STREAM: non-temporal stores (logits: write-once, keep L2
// for weights).  64x64 block tile, 8 waves each a 16x32 strip, BK=32,
// double-buffered LDS fed by async global->LDS DMA (one barrier per stage).
template <int EPI, int STREAM>
__global__ __launch_bounds__(256) void gemm_wmma(
    const float* __restrict__ A, const float* __restrict__ Bm,
    const float* __restrict__ bias, const float* __restrict__ Res,
    float* __restrict__ C, int M, int N, int K) {
  __shared__ float As[2][64][36]; // stride 36: conflict-free A-frag reads,
                                  // 144B row pitch keeps 16B-aligned DMA
  __shared__ float Bs[2][32][80]; // stride 80: conflict-free incl. kb offset

  const int tid = threadIdx.x;
  const int lane = tid & 31;
  const int w = tid >> 5;
  const int nA = lane & 15;
  const int kb = (lane >> 4) * 2;
  const int mt = (w & 3) * 16;  // wave's M tile inside block
  const int nt = (w >> 2) * 32; // wave's N offset inside block
  const int mb = blockIdx.y * 64;
  const int nb = blockIdx.x * 64;

  v8f c0 = zero8();
  v8f c1 = zero8();

  // staging roles: A panel 64x32 = 512 float4 (2/thread), B panel 32x64 same
  const int ar0 = tid >> 3;           // 0..31
  const int akq = (tid & 7) * 4;      // 0..28
  const int br0 = tid >> 4;           // 0..15
  const int bc4 = (tid & 15) * 4;     // 0..60

  uint64_t ga0 = (uint64_t)(uintptr_t)(A + (size_t)(mb + ar0) * K + akq);
  uint64_t ga1 = (uint64_t)(uintptr_t)(A + (size_t)(mb + ar0 + 32) * K + akq);
  uint64_t gb0 = (uint64_t)(uintptr_t)(Bm + (size_t)br0 * N + nb + bc4);
  uint64_t gb1 = (uint64_t)(uintptr_t)(Bm + (size_t)(br0 + 16) * N + nb + bc4);
  const uint64_t aStep = 32u * 4u;            // BK floats
  const uint64_t bStep = (uint64_t)32 * N * 4; // BK rows

  unsigned lA0[2], lA1[2], lB0[2], lB1[2];
#pragma unroll
  for (int b = 0; b < 2; ++b) {
    lA0[b] = lds_off(&As[b][ar0][akq]);
    lA1[b] = lds_off(&As[b][ar0 + 32][akq]);
    lB0[b] = lds_off(&Bs[b][br0][bc4]);
    lB1[b] = lds_off(&Bs[b][br0 + 16][bc4]);
  }

  // prologue: stage 0 into buffer 0
  ASYNC_B128(lA0[0], ga0);
  ASYNC_B128(lA1[0], ga1);
  ASYNC_B128(lB0[0], gb0);
  ASYNC_B128(lB1[0], gb1);
  ga0 += aStep; ga1 += aStep; gb0 += bStep; gb1 += bStep;
  WAIT_ASYNC();
  __syncthreads();

  const int nStage = K >> 5;
  for (int st = 0; st < nStage; ++st) {
    const int cur = st & 1;
    if (st + 1 < nStage) { // kick DMA for next stage into other buffer
      const int nxt = cur ^ 1;
      ASYNC_B128(lA0[nxt], ga0);
      ASYNC_B128(lA1[nxt], ga1);
      ASYNC_B128(lB0[nxt], gb0);
      ASYNC_B128(lB1[nxt], gb1);
      ga0 += aStep; ga1 += aStep; gb0 += bStep; gb1 += bStep;
    }
#pragma unroll
    for (int kk = 0; kk < 8; ++kk) { // 16 WMMAs overlapping the DMA
      const int k4 = kk * 4;
      v2f a, b0, b1;
      a.x = As[cur][mt + nA][k4 + kb];
      a.y = As[cur][mt + nA][k4 + kb + 1];
      b0.x = Bs[cur][k4 + kb][nt + nA];
      b0.y = Bs[cur][k4 + kb + 1][nt + nA];
      b1.x = Bs[cur][k4 + kb][nt + 16 + nA];
      b1.y = Bs[cur][k4 + kb + 1][nt + 16 + nA];
      c0 = wmma4(a, b0, c0);
      c1 = wmma4(a, b1, c1);
    }
    WAIT_ASYNC();
    __syncthreads();
  }

  const int half = lane >> 4;
#pragma unroll
  for (int r = 0; r < 8; ++r) {
    const int row = mb + mt + r + half * 8;
    const int col0 = nb + nt + nA;
    const int col1 = col0 + 16;
    float v0 = c0[r] + (bias ? bias[col0] : 0.0f);
    float v1 = c1[r] + (bias ? bias[col1] : 0.0f);
    if (EPI == 1) { // exact erf-GELU
      v0 = 0.5f * v0 * (1.0f + erff(v0 * 0.70710678118654752f));
      v1 = 0.5f * v1 * (1.0f + erff(v1 * 0.70710678118654752f));
    } else if (EPI == 2) {
      v0 += Res[(size_t)row * N + col0];
      v1 += Res[(size_t)row * N + col1];
    }
    if (STREAM) {
      __builtin_nontemporal_store(v0, &C[(size_t)row * N + col0]);
      __builtin_nontemporal_store(v1, &C[(size_t)row * N + col1]);
    } else {
      C[(size_t)row * N + col0] = v0;
      C[(size_t)row * N + col1] = v1;
    }
  }
}

// ---------------------------------------------------------------- attention
// Flash attention, causal. qkv rows of 1536: [Q|K|V] each 512 = 8 heads x 64.
// Block = 128 queries of one (b,h); 8 waves, 16 q-rows each.
__global__ __launch_bounds__(256) void attn_flash(const float* __restrict__ qkv,
                                                  float* __restrict__ out) {
  __shared__ float Ks[16][68];    // [key][d], stride 68 conflict-free
  __shared__ float Vs[16][80];    // [key][d], stride 80 conflict-free
  __shared__ float Ps[8][16][20]; // per-wave P tile [q][key], stride 20

  const int tid = threadIdx.x;
  const int lane = tid & 31;
  const int w = tid >> 5;
  const int half = lane >> 4;
  const int nA = lane & 15;
  const int kb = half * 2;

  const int qt = blockIdx.x & 15; // 16 q-tiles of 128
  const int bh = blockIdx.x >> 4;
  const int bb = bh >> 3;
  const int hh = bh & 7;
  const int qbase = qt * 128;
  const int qrow = qbase + w * 16;

  // Q fragments (pre-scaled by 1/sqrt(D)=0.125), kept in registers
  v2f qf[16];
  {
    const float* qp = qkv + (size_t)(bb * SSQ + qrow + nA) * (3 * EE) + hh * DD;
#pragma unroll
    for (int kk = 0; kk < 16; ++kk) {
      const int kd = kk * 4 + kb;
      qf[kk].x = qp[kd] * 0.125f;
      qf[kk].y = qp[kd + 1] * 0.125f;
    }
  }

  v8f o0 = zero8(), o1 = zero8(), o2 = zero8(), o3 = zero8();
  float mrow[8], lrow[8];
#pragma unroll
  for (int r = 0; r < 8; ++r) {
    mrow[r] = -3.0e38f;
    lrow[r] = 0.0f;
  }

  // async staging addresses for K/V tiles (16 keys x 64 d, one b128 per thread)
  const int kr = tid >> 4;
  const int c4 = (tid & 15) * 4;
  uint64_t gk = (uint64_t)(uintptr_t)(qkv + (size_t)(bb * SSQ + kr) * (3 * EE) +
                                      hh * DD + EE + c4);
  const uint64_t kStep = (uint64_t)16 * (3 * EE) * 4; // 16 rows per tile
  const unsigned lK = lds_off(&Ks[kr][c4]);
  const unsigned lV = lds_off(&Vs[kr][c4]);

  const int ktmax = qt * 8 + 7; // last key tile needed for q <= qbase+127
  for (int kt = 0; kt <= ktmax; ++kt) {
    const int k0 = kt * 16;
    ASYNC_B128(lK, gk);            // K tile
    ASYNC_B128(lV, gk + EE * 4);   // V tile (+512 floats)
    gk += kStep;
    WAIT_ASYNC();
    __syncthreads();

    // S = (Q/sqrt(D)) @ K^T : 16 WMMA k-steps over D=64
    v8f s = zero8();
#pragma unroll
    for (int kk = 0; kk < 16; ++kk) {
      const int k4 = kk * 4;
      v2f bk;
      bk.x = Ks[nA][k4 + kb];
      bk.y = Ks[nA][k4 + kb + 1];
      s = wmma4(qf[kk], bk, s);
    }

    // online softmax; row = (r + 8*half) spread over 16 lanes (= 16 keys)
    const int kidx = k0 + nA;
#pragma unroll
    for (int r = 0; r < 8; ++r) {
      const int qidx = qrow + r + half * 8;
      float sv = (kidx <= qidx) ? s[r] : -3.0e38f;
      float mx = sv;
      mx = fmaxf(mx, __shfl_xor(mx, 1));
      mx = fmaxf(mx, __shfl_xor(mx, 2));
      mx = fmaxf(mx, __shfl_xor(mx, 4));
      mx = fmaxf(mx, __shfl_xor(mx, 8));
      const float mnew = fmaxf(mrow[r], mx);
      const float alpha = expf(mrow[r] - mnew);
      const float pv = expf(sv - mnew); // masked lanes underflow to 0
      float sum = pv;
      sum += __shfl_xor(sum, 1);
      sum += __shfl_xor(sum, 2);
      sum += __shfl_xor(sum, 4);
      sum += __shfl_xor(sum, 8);
      lrow[r] = lrow[r] * alpha + sum;
      mrow[r] = mnew;
      o0[r] *= alpha;
      o1[r] *= alpha;
      o2[r] *= alpha;
      o3[r] *= alpha;
      Ps[w][r + half * 8][nA] = pv; // C-layout -> LDS for A-layout reload
    }
    asm volatile("" ::: "memory"); // keep P stores before P reloads
                                   // (per-wave LDS ops are in order in HW)

    // O += P @ V : K=16 keys -> 4 WMMA k-steps, 4 N-tiles of D
#pragma unroll
    for (int kk = 0; kk < 4; ++kk) {
      const int k4 = kk * 4;
      v2f a, b;
      a.x = Ps[w][nA][k4 + kb];
      a.y = Ps[w][nA][k4 + kb + 1];
      b.x = Vs[k4 + kb][0 + nA];
      b.y = Vs[k4 + kb + 1][0 + nA];
      o0 = wmma4(a, b, o0);
      b.x = Vs[k4 + kb][16 + nA];
      b.y = Vs[k4 + kb + 1][16 + nA];
      o1 = wmma4(a, b, o1);
      b.x = Vs[k4 + kb][32 + nA];
      b.y = Vs[k4 + kb + 1][32 + nA];
      o2 = wmma4(a, b, o2);
      b.x = Vs[k4 + kb][48 + nA];
      b.y = Vs[k4 + kb + 1][48 + nA];
      o3 = wmma4(a, b, o3);
    }
    __syncthreads(); // Ks/Vs reused next tile
  }

#pragma unroll
  for (int r = 0; r < 8; ++r) {
    const float inv = 1.0f / lrow[r];
    float* op = out + (size_t)(bb * SSQ + qrow + r + half * 8) * EE + hh * DD;
    op[0 + nA] = o0[r] * inv;
    op[16 + nA] = o1[r] * inv;
    op[32 + nA] = o2[r] * inv;
    op[48 + nA] = o3[r] * inv;
  }
}

// ---------------------------------------------------------------- launch
extern "C" void kernel_launch(void* const* d_in, const int* in_sizes, int n_in,
                              void* d_out, int out_size, void* d_ws,
                              size_t ws_size, hipStream_t stream) {
  (void)in_sizes;
  (void)n_in;
  (void)out_size;
  (void)ws_size;
  const int* ids = (const int*)d_in[0];
  const float* tok = (const float*)d_in[1];
  const float* pos = (const float*)d_in[2];
  const float* qkv_w = (const float*)d_in[3];
  const float* qkv_b = (const float*)d_in[4];
  const float* proj_w = (const float*)d_in[5];
  const float* proj_b = (const float*)d_in[6];
  const float* fc1_w = (const float*)d_in[7];
  const float* fc1_b = (const float*)d_in[8];
  const float* fc2_w = (const float*)d_in[9];
  const float* fc2_b = (const float*)d_in[10];
  const float* ln1_g = (const float*)d_in[11];
  const float* ln2_g = (const float*)d_in[12];
  const float* lnf_g = (const float*)d_in[13];
  const float* out_w = (const float*)d_in[14];
  float* logits = (float*)d_out;

  float* x = (float*)d_ws;                 // [4096,512]
  float* h = x + (size_t)MM * EE;          // [4096,512]  LN output
  float* qkvb = h + (size_t)MM * EE;       // [4096,1536]
  float* att = qkvb + (size_t)MM * 3 * EE; // [4096,512]
  float* mid = att + (size_t)MM * EE;      // [4096,2048]

  embed_k<<<(MM * EE) / 256, 256, 0, stream>>>(ids, tok, pos, x);

  for (int l = 0; l < LL; ++l) {
    ln_rows<<<MM / 8, 256, 0, stream>>>(x, ln1_g + l * EE, h);
    gemm_wmma<0, 0><<<dim3((3 * EE) / 64, MM / 64), 256, 0, stream>>>(
        h, qkv_w + (size_t)l * EE * 3 * EE, qkv_b + l * 3 * EE, nullptr, qkvb,
        MM, 3 * EE, EE);
    attn_flash<<<(BBB * HH) * (SSQ / 128), 256, 0, stream>>>(qkvb, att);
    gemm_wmma<2, 0><<<dim3(EE / 64, MM / 64), 256, 0, stream>>>(
        att, proj_w + (size_t)l * EE * EE, proj_b + l * EE, x, x, MM, EE, EE);
    ln_rows<<<MM / 8, 256, 0, stream>>>(x, ln2_g + l * EE, h);
    gemm_wmma<1, 0><<<dim3((4 * EE) / 64, MM / 64), 256, 0, stream>>>(
        h, fc1_w + (size_t)l * EE * 4 * EE, fc1_b + l * 4 * EE, nullptr, mid,
        MM, 4 * EE, EE);
    gemm_wmma<2, 0><<<dim3(EE / 64, MM / 64), 256, 0, stream>>>(
        mid, fc2_w + (size_t)l * 4 * EE * EE, fc2_b + l * EE, x, x, MM, EE,
        4 * EE);
  }
  ln_rows<<<MM / 8, 256, 0, stream>>>(x, lnf_g, h);
  gemm_wmma<0, 1><<<dim3(VV / 64, MM / 64), 256, 0, stream>>>(
      h, out_w, nullptr, nullptr, logits, MM, VV, EE);
}